// TrainableBilateralFilter_88064009437990
// MI455X (gfx1250) — compile-verified
//
#include <hip/hip_runtime.h>
#include <hip/hip_bf16.h>
#include <stdint.h>

#define HALF 4
#define TILE 16
#define SPAN (TILE + 2 * HALF)   // 24
#define NTHREADS 256

// Accumulate 81 taps for one pixel. CHECK=true adds per-tap bounds masking
// (weight forced to 0 for out-of-image neighbors, matching the reference's
// zero-padded validity mask). LDS holds clamped (finite) values so w=0 never
// multiplies Inf/NaN.
template <bool CHECK>
__device__ __forceinline__ void accum_taps(const float* tile, int lx, int ly,
                                           int gx, int gy, int W, int H,
                                           float cx, float cy, float cr,
                                           float center, float& num, float& den) {
#pragma unroll
  for (int dy = -HALF; dy <= HALF; ++dy) {
#pragma unroll
    for (int dx = -HALF; dx <= HALF; ++dx) {
      float nb = tile[(ly + HALF + dy) * SPAN + (lx + HALF + dx)];
      float d = center - nb;
      // log2-domain: exp2(-(dx^2)*cx - (dy^2)*cy - d*d*cr)
      float sp = __builtin_fmaf((float)(-(dx * dx)), cx,
                                (float)(-(dy * dy)) * cy);
      float arg = __builtin_fmaf(d * d, -cr, sp);
      float w = __builtin_amdgcn_exp2f(arg);   // v_exp_f32 (TRANS, co-executes)
      if (CHECK) {
        bool ok = ((unsigned)(gx + dx) < (unsigned)W) &&
                  ((unsigned)(gy + dy) < (unsigned)H);
        w = ok ? w : 0.0f;
      }
      num = __builtin_fmaf(w, nb, num);
      den += w;
    }
  }
}

__global__ __launch_bounds__(NTHREADS)
void bilateral_kernel(const float* __restrict__ x,
                      const float* __restrict__ psx,
                      const float* __restrict__ psy,
                      const float* __restrict__ psr,
                      float* __restrict__ out, int H, int W) {
  __shared__ float tile[SPAN * SPAN];

  const int tid = threadIdx.x;
  const int bx = blockIdx.x * TILE;
  const int by = blockIdx.y * TILE;
  const int ibase = blockIdx.z * H * W;   // <= 16*1024*1024, fits int32

  // ---- Stage 24x24 halo tile: async global -> LDS (gfx1250 path) ----
  // Per-lane gather with coordinates clamped into the image so LDS content is
  // always a real (finite) pixel; validity is enforced at compute time.
  const uint32_t lds_base = (uint32_t)(uintptr_t)&tile[0];  // low 32b of generic LDS ptr = LDS offset
  for (int i = tid; i < SPAN * SPAN; i += NTHREADS) {
    int ly = i / SPAN;
    int lx = i - ly * SPAN;
    int gy = by - HALF + ly;
    int gx = bx - HALF + lx;
    gy = min(max(gy, 0), H - 1);
    gx = min(max(gx, 0), W - 1);
    uint32_t goff = (uint32_t)((ibase + gy * W + gx) * 4);  // byte offset, < 2^31
    uint32_t laddr = lds_base + (uint32_t)i * 4u;
    asm volatile("global_load_async_to_lds_b32 %0, %1, %2"
                 :
                 : "v"(laddr), "v"(goff), "s"(x)
                 : "memory");
  }
  asm volatile("s_wait_asynccnt 0" ::: "memory");  // my wave's ASYNCcnt drained
  __syncthreads();                                  // tile visible to workgroup

  // ---- Per-thread constants (scalars live in device memory) ----
  const float log2e = 1.4426950408889634f;
  const float sxv = psx[0];
  const float syv = psy[0];
  const float srv = psr[0];
  const float cx = (0.5f / (sxv * sxv)) * log2e;
  const float cy = (0.5f / (syv * syv)) * log2e;
  const float cr = (0.5f / (srv * srv)) * log2e;

  const int lx = tid & (TILE - 1);
  const int ly = tid >> 4;
  const int gx = bx + lx;
  const int gy = by + ly;

  const float center = tile[(ly + HALF) * SPAN + (lx + HALF)];
  float num = 0.0f, den = 0.0f;

  // Uniform (per-block) branch: interior tiles skip all bounds masking.
  const bool interior = (bx >= HALF) && (by >= HALF) &&
                        (bx + TILE + HALF <= W) && (by + TILE + HALF <= H);
  if (interior) {
    accum_taps<false>(tile, lx, ly, gx, gy, W, H, cx, cy, cr, center, num, den);
  } else {
    accum_taps<true>(tile, lx, ly, gx, gy, W, H, cx, cy, cr, center, num, den);
  }

  // den >= 1 (center tap weight is exp2(0) == 1), so division is safe.
  out[ibase + gy * W + gx] = num / den;
}

extern "C" void kernel_launch(void* const* d_in, const int* in_sizes, int n_in,
                              void* d_out, int out_size, void* d_ws, size_t ws_size,
                              hipStream_t stream) {
  const float* x   = (const float*)d_in[0];
  const float* psx = (const float*)d_in[1];
  const float* psy = (const float*)d_in[2];
  const float* psr = (const float*)d_in[3];
  float* out = (float*)d_out;

  const int H = 1024, W = 1024;
  const int B = in_sizes[0] / (H * W);   // 16

  dim3 grid(W / TILE, H / TILE, B);      // 64 x 64 x 16 blocks
  dim3 block(NTHREADS);                  // 8 wave32s per block
  bilateral_kernel<<<grid, block, 0, stream>>>(x, psx, psy, psr, out, H, W);
}